// WeightLinear_46505905881421
// MI455X (gfx1250) — compile-verified
//
#include <hip/hip_runtime.h>
#include <hip/hip_bf16.h>

typedef __attribute__((ext_vector_type(2))) float v2f;
typedef __attribute__((ext_vector_type(8))) float v8f;

#define BATCH 32768
#define IN_F  34
#define OUT_F 1024

// One wave32 computes one 16x16 fp32 output tile via V_WMMA_F32_16X16X4_F32.
// Block = 256 threads = 8 waves = 8 consecutive M-tiles at a single N-tile
// (keeps the 16 weight columns hot in L1 across the block's waves).
//
// K = 34 -> 8 unconditional K=4 steps (k=0..31) + 1 branch-free tail step
// covering k=32,33 (k=34,35 are zero padding, realized as v_cndmask zeros on
// the upper half-wave's fragments -- no EXEC manipulation, WMMA-legal).
__global__ __launch_bounds__(256) void WeightLinear_46505905881421_kernel(
    const float* __restrict__ score,   // [BATCH][IN_F]
    const float* __restrict__ x,       // [BATCH][IN_F]
    const float* __restrict__ weight,  // [IN_F][OUT_F]
    const float* __restrict__ bias,    // [OUT_F]
    float* __restrict__ out)           // [BATCH][OUT_F]
{
    const int lane = threadIdx.x & 31;
    const int wave = threadIdx.x >> 5;

    const int n0 = blockIdx.x * 16;                  // N tile (64 tiles)
    const int m0 = (blockIdx.y * 8 + wave) * 16;     // M tile (2048 tiles)

    const int half = lane >> 4;    // 0: lanes 0-15, 1: lanes 16-31
    const int l16  = lane & 15;

    const int m = m0 + l16;        // A-fragment row owned by this lane
    const int n = n0 + l16;        // B/C-fragment column owned by this lane

    const float* __restrict__ xrow = x + m * IN_F;
    const float* __restrict__ srow = score + m * IN_F;

    v8f acc = {};

    // Main body: k = 0..31, ka = k0 + 2*half <= 30 < 34 -> no bounds checks.
    // A layout (16x4 f32): VGPR0 = K {0|2}, VGPR1 = K {1|3} selected by half.
    // B layout (4x16 f32): same K split, N striped across lanes 0-15.
    #pragma unroll
    for (int k0 = 0; k0 < 32; k0 += 4) {
        const int ka = k0 + half * 2;

        v2f a;
        a.x = xrow[ka]     * srow[ka];
        a.y = xrow[ka + 1] * srow[ka + 1];

        v2f b;
        b.x = weight[ka * OUT_F + n];
        b.y = weight[(ka + 1) * OUT_F + n];

        // 8 args: (neg_a, A, neg_b, B, c_mod, C, reuse_a, reuse_b)
        acc = __builtin_amdgcn_wmma_f32_16x16x4_f32(
            false, a, false, b, (short)0, acc, false, false);
    }

    // Tail step: logical K = 32..35; only 32,33 exist. Lanes 0-15 own K=32,33
    // (valid); lanes 16-31 own K=34,35 (padding -> zero via cndmask). Load
    // address is the constant 32/33 for all lanes: unconditional + in-bounds.
    {
        const float a0 = xrow[32] * srow[32];
        const float a1 = xrow[33] * srow[33];
        const float b0 = weight[32 * OUT_F + n];
        const float b1 = weight[33 * OUT_F + n];
        const bool lo = (half == 0);

        v2f a; a.x = lo ? a0 : 0.0f; a.y = lo ? a1 : 0.0f;
        v2f b; b.x = lo ? b0 : 0.0f; b.y = lo ? b1 : 0.0f;

        acc = __builtin_amdgcn_wmma_f32_16x16x4_f32(
            false, a, false, b, (short)0, acc, false, false);
    }

    // C/D layout: VGPR r holds (M = m0 + 8*half + r, N = n0 + l16).
    // Each store coalesces into two contiguous 64B segments per half-wave.
    const float bv = bias[n];
    float* __restrict__ orow = out + (m0 + half * 8) * OUT_F + n;
    #pragma unroll
    for (int r = 0; r < 8; ++r) {
        orow[r * OUT_F] = acc[r] + bv;
    }
}

extern "C" void kernel_launch(void* const* d_in, const int* in_sizes, int n_in,
                              void* d_out, int out_size, void* d_ws, size_t ws_size,
                              hipStream_t stream) {
    (void)in_sizes; (void)n_in; (void)d_ws; (void)ws_size; (void)out_size;

    const float* score  = (const float*)d_in[0];   // (32768, 34)
    const float* x      = (const float*)d_in[1];   // (32768, 34)
    const float* weight = (const float*)d_in[2];   // (1, 34, 1024)
    const float* bias   = (const float*)d_in[3];   // (1, 1024)
    float* out          = (float*)d_out;           // (32768, 1024)

    dim3 grid(OUT_F / 16, BATCH / (16 * 8));       // (64, 256)
    dim3 block(256);
    WeightLinear_46505905881421_kernel<<<grid, block, 0, stream>>>(
        score, x, weight, bias, out);
}